// FarmGNN_67740224193326
// MI455X (gfx1250) — compile-verified
//
#include <hip/hip_runtime.h>
#include <hip/hip_bf16.h>

#define NEG_SLOPE 0.2f
#define LN_EPS 1e-5f

typedef float v2f __attribute__((ext_vector_type(2)));
typedef float v8f __attribute__((ext_vector_type(8)));

// ---------------------------------------------------------------------------
// Utility device helpers
// ---------------------------------------------------------------------------
__device__ __forceinline__ float wave_sum32(float v) {
#pragma unroll
    for (int off = 16; off > 0; off >>= 1) v += __shfl_down(v, off, 32);
    return v;
}

// float atomic max via int/uint ordering trick (valid with -inf init)
__device__ __forceinline__ void atomicMaxF(float* addr, float val) {
    if (val >= 0.0f) {
        atomicMax((int*)addr, __float_as_int(val));
    } else {
        atomicMin((unsigned int*)addr, __float_as_uint(val));
    }
}

// ---------------------------------------------------------------------------
// Fill
// ---------------------------------------------------------------------------
__global__ void fill_kernel(float* __restrict__ p, float v, long n) {
    long i = (long)blockIdx.x * blockDim.x + threadIdx.x;
    if (i < n) p[i] = v;
}

// ---------------------------------------------------------------------------
// Degree + self-loop edge_attr (scatter-mean of edge_attr onto dst)
// ---------------------------------------------------------------------------
__global__ void deg_loop_kernel(const int* __restrict__ ei, const float* __restrict__ ea,
                                float* __restrict__ deg, float* __restrict__ lsum, int E) {
    int e = blockIdx.x * blockDim.x + threadIdx.x;
    if (e < E) {
        int d = ei[E + e];
        atomicAdd(&deg[d], 1.0f);
        atomicAdd(&lsum[d], ea[e]);
    }
}

__global__ void loop_div_kernel(float* __restrict__ la, const float* __restrict__ deg, int Nn) {
    int n = blockIdx.x * blockDim.x + threadIdx.x;
    if (n < Nn) la[n] = la[n] / fmaxf(deg[n], 1.0f);
}

// ---------------------------------------------------------------------------
// WMMA f32 GEMM: C[M x N] = A[M x K] * B[K x N]   (row major, N%64==0, K%4==0)
// One wave32 per 16x64 C strip (4 accumulators): A 16x4 is loaded once per
// k-step (one b64 per lane) and feeds 4 back-to-back V_WMMA_F32_16X16X4_F32.
// A 16x4 layout : lanes 0-15 row M=l, VGPR0/1 = K0,K1 ; lanes 16-31 = K2,K3
// B 4x16 layout : lanes 0-15 col N=l, VGPR0/1 = K0,K1 ; lanes 16-31 = K2,K3
// C 16x16 layout: VGPR r -> row r (lanes 0-15) / row r+8 (lanes 16-31)
// ---------------------------------------------------------------------------
__global__ void gemm_wmma_f32(const float* __restrict__ A, const float* __restrict__ B,
                              float* __restrict__ C, int M, int N, int K) {
    int wave    = (blockIdx.x * blockDim.x + threadIdx.x) >> 5;
    int ngrp    = N >> 6;                 // 64-column groups
    int tiles_m = (M + 15) >> 4;
    int tm = wave / ngrp;
    int tg = wave % ngrp;
    if (tm >= tiles_m) return;

    int lane = threadIdx.x & 31;
    int half = lane >> 4;   // 0: lanes 0-15 (K0,K1), 1: lanes 16-31 (K2,K3)
    int l    = lane & 15;

    int arow   = tm * 16 + l;
    int arow_c = arow < M ? arow : M - 1;               // clamp loads; stores guarded
    const float* Ap = A + (size_t)arow_c * K + 2 * half; // per-lane A base (K0/K2)
    const float* Bp = B + (size_t)(2 * half) * N + tg * 64 + l; // per-lane B base

    v8f acc0 = {}, acc1 = {}, acc2 = {}, acc3 = {};
    for (int k0 = 0; k0 < K; k0 += 4) {
        v2f a = *(const v2f*)(Ap + k0);                 // one b64: K{0,1} or K{2,3}
        const float* Br0 = Bp + (size_t)k0 * N;
        const float* Br1 = Br0 + N;
        v2f b0, b1, b2, b3;
        b0.x = Br0[0];  b0.y = Br1[0];
        b1.x = Br0[16]; b1.y = Br1[16];
        b2.x = Br0[32]; b2.y = Br1[32];
        b3.x = Br0[48]; b3.y = Br1[48];
        acc0 = __builtin_amdgcn_wmma_f32_16x16x4_f32(false, a, false, b0, (short)0, acc0, false, false);
        acc1 = __builtin_amdgcn_wmma_f32_16x16x4_f32(false, a, false, b1, (short)0, acc1, false, false);
        acc2 = __builtin_amdgcn_wmma_f32_16x16x4_f32(false, a, false, b2, (short)0, acc2, false, false);
        acc3 = __builtin_amdgcn_wmma_f32_16x16x4_f32(false, a, false, b3, (short)0, acc3, false, false);
    }

#pragma unroll
    for (int r = 0; r < 8; ++r) {
        int row = tm * 16 + r + 8 * half;
        if (row < M) {
            float* Cr = C + (size_t)row * N + tg * 64 + l;
            Cr[0]  = acc0[r];
            Cr[16] = acc1[r];
            Cr[32] = acc2[r];
            Cr[48] = acc3[r];
        }
    }
}

// ---------------------------------------------------------------------------
// Edge pass A: raw attention logits + per-(dst,head) running max.
// One wave per edge (E real edges + Nn self-loops).
// ---------------------------------------------------------------------------
template <int H, int C>
__global__ void edge_alpha_kernel(const int* __restrict__ ei, const float* __restrict__ ea,
                                  const float* __restrict__ la,
                                  const float* __restrict__ xl, const float* __restrict__ xr,
                                  const float* __restrict__ We, const float* __restrict__ att,
                                  float* __restrict__ alpha, float* __restrict__ amax,
                                  int E, int Nn) {
    constexpr int HC  = H * C;
    constexpr int KPL = HC / 32;
    int e = blockIdx.x * (blockDim.x >> 5) + (threadIdx.x >> 5);
    if (e >= E + Nn) return;
    int lane = threadIdx.x & 31;

    int src, dst; float eav;
    if (e < E) { src = ei[e]; dst = ei[E + e]; eav = ea[e]; }
    else       { src = e - E; dst = src;       eav = la[src]; }

    const float* pl = xl + (size_t)src * HC;
    const float* pr = xr + (size_t)dst * HC;

    float acc[H];
#pragma unroll
    for (int h = 0; h < H; ++h) acc[h] = 0.0f;

#pragma unroll
    for (int k = 0; k < KPL; ++k) {
        int ch = lane + 32 * k;
        float m = pl[ch] + pr[ch] + eav * We[ch];
        m = m > 0.0f ? m : NEG_SLOPE * m;
        acc[(32 * k) / C] += m * att[ch];
    }

#pragma unroll
    for (int h = 0; h < H; ++h) {
        float v = wave_sum32(acc[h]);
        if (lane == 0) {
            alpha[(size_t)e * H + h] = v;
            atomicMaxF(&amax[(size_t)dst * H + h], v);
        }
    }
}

// ---------------------------------------------------------------------------
// Edge pass B: w = exp(alpha - amax[dst]); accumulate denom and w*xl[src].
// ---------------------------------------------------------------------------
template <int H, int C>
__global__ void edge_agg_kernel(const int* __restrict__ ei,
                                const float* __restrict__ alpha, const float* __restrict__ amax,
                                const float* __restrict__ xl,
                                float* __restrict__ agg, float* __restrict__ denom,
                                int E, int Nn) {
    constexpr int HC  = H * C;
    constexpr int KPL = HC / 32;
    int e = blockIdx.x * (blockDim.x >> 5) + (threadIdx.x >> 5);
    if (e >= E + Nn) return;
    int lane = threadIdx.x & 31;

    int src, dst;
    if (e < E) { src = ei[e]; dst = ei[E + e]; }
    else       { src = e - E; dst = src; }

    float w[H];
#pragma unroll
    for (int h = 0; h < H; ++h)
        w[h] = expf(alpha[(size_t)e * H + h] - amax[(size_t)dst * H + h]);

    const float* pl = xl  + (size_t)src * HC;
    float*       pd = agg + (size_t)dst * HC;
#pragma unroll
    for (int k = 0; k < KPL; ++k) {
        int ch = lane + 32 * k;
        atomicAdd(&pd[ch], w[(32 * k) / C] * pl[ch]);
    }
    if (lane < H) atomicAdd(&denom[(size_t)dst * H + lane], w[lane]);
}

// ---------------------------------------------------------------------------
// Node finalize: out = agg/denom + bias -> LayerNorm -> ELU (in place).
// Optionally fused graph mean-pool accumulation (layer 2).
// One block of HC threads per node.
// ---------------------------------------------------------------------------
template <int H, int HC>
__global__ void finalize_kernel(float* __restrict__ agg, const float* __restrict__ denom,
                                const float* __restrict__ bias,
                                const float* __restrict__ g, const float* __restrict__ beta,
                                int Nn, const int* __restrict__ batch,
                                float* __restrict__ pool, float* __restrict__ cnt, int do_pool) {
    constexpr int C  = HC / H;
    constexpr int NW = HC / 32;
    int n = blockIdx.x;
    int c = threadIdx.x;
    if (n >= Nn) return;

    float v = agg[(size_t)n * HC + c] / (denom[(size_t)n * H + c / C] + 1e-16f) + bias[c];

    __shared__ float sred[NW];
    __shared__ float smu, svar;
    int wid  = threadIdx.x >> 5;
    int lane = threadIdx.x & 31;

    float s = wave_sum32(v);
    if (lane == 0) sred[wid] = s;
    __syncthreads();
    if (threadIdx.x == 0) {
        float t = 0.0f;
#pragma unroll
        for (int wv = 0; wv < NW; ++wv) t += sred[wv];
        smu = t / (float)HC;
    }
    __syncthreads();
    float mu = smu;
    float d  = v - mu;

    s = wave_sum32(d * d);
    if (lane == 0) sred[wid] = s;
    __syncthreads();
    if (threadIdx.x == 0) {
        float t = 0.0f;
#pragma unroll
        for (int wv = 0; wv < NW; ++wv) t += sred[wv];
        svar = t / (float)HC;
    }
    __syncthreads();

    float y = d * rsqrtf(svar + LN_EPS) * g[c] + beta[c];
    y = y > 0.0f ? y : (expf(y) - 1.0f);   // ELU
    agg[(size_t)n * HC + c] = y;

    if (do_pool) {
        int gb = batch[n];
        atomicAdd(&pool[(size_t)gb * HC + c], y);
        if (c == 0) atomicAdd(&cnt[gb], 1.0f);
    }
}

// ---------------------------------------------------------------------------
// Final mean-pool divide -> d_out (64 x 128)
// ---------------------------------------------------------------------------
__global__ void pool_div_kernel(const float* __restrict__ pool, const float* __restrict__ cnt,
                                float* __restrict__ out, int n) {
    int i = blockIdx.x * blockDim.x + threadIdx.x;
    if (i < n) out[i] = pool[i] / fmaxf(cnt[i >> 7], 1.0f);
}

// ---------------------------------------------------------------------------
// Host launch
// ---------------------------------------------------------------------------
extern "C" void kernel_launch(void* const* d_in, const int* in_sizes, int n_in,
                              void* d_out, int out_size, void* d_ws, size_t ws_size,
                              hipStream_t stream) {
    const float* x     = (const float*)d_in[0];
    const int*   ei    = (const int*)d_in[1];
    const float* ea    = (const float*)d_in[2];
    const int*   batch = (const int*)d_in[3];
    const float* Wl1   = (const float*)d_in[4];
    const float* Wr1   = (const float*)d_in[5];
    const float* We1   = (const float*)d_in[6];
    const float* att1  = (const float*)d_in[7];
    const float* b1    = (const float*)d_in[8];
    const float* g1    = (const float*)d_in[9];
    const float* be1   = (const float*)d_in[10];
    const float* Wl2   = (const float*)d_in[11];
    const float* Wr2   = (const float*)d_in[12];
    const float* We2   = (const float*)d_in[13];
    const float* att2  = (const float*)d_in[14];
    const float* b2    = (const float*)d_in[15];
    const float* g2    = (const float*)d_in[16];
    const float* be2   = (const float*)d_in[17];

    const int Nn = in_sizes[0] / 8;   // 50000
    const int E  = in_sizes[1] / 2;   // 800000
    const int E2 = E + Nn;
    const int NG = 64;

    // workspace layout (floats)
    float* ws = (float*)d_ws;
    float* XL = ws;                               // Nn*256  (xl1 / xl2)
    float* XR = XL + (size_t)Nn * 256;            // Nn*256  (xr1 / xr2)
    float* AG = XR + (size_t)Nn * 256;            // Nn*256  (agg1/h1, then agg2/h2)
    float* AL = AG + (size_t)Nn * 256;            // E2*4    (alpha)
    float* AM = AL + (size_t)E2 * 4;              // Nn*4    (segment max)
    float* DN = AM + (size_t)Nn * 4;              // Nn*4    (softmax denom)
    float* LA = DN + (size_t)Nn * 4;              // Nn      (self-loop attr)
    float* DG = LA + Nn;                          // Nn      (degree)
    float* PO = DG + Nn;                          // 64*128  (pool accum)
    float* CT = PO + (size_t)NG * 128;            // 64      (graph counts)

    const float NINF = -__builtin_huge_valf();
#define FILLF(p, v, n) fill_kernel<<<(int)(((long)(n) + 255) / 256), 256, 0, stream>>>((p), (v), (long)(n))

    // ---- init ----
    FILLF(DG, 0.0f, Nn);
    FILLF(LA, 0.0f, Nn);
    FILLF(AM, NINF, (long)Nn * 4);
    FILLF(DN, 0.0f, (long)Nn * 4);
    FILLF(AG, 0.0f, (long)Nn * 256);
    FILLF(PO, 0.0f, (long)NG * 128);
    FILLF(CT, 0.0f, NG);

    // ---- degree + self-loop edge_attr mean ----
    deg_loop_kernel<<<(E + 255) / 256, 256, 0, stream>>>(ei, ea, DG, LA, E);
    loop_div_kernel<<<(Nn + 255) / 256, 256, 0, stream>>>(LA, DG, Nn);

    // ---- layer 1 node transforms via WMMA f32 (M=Nn, N=256, K=8) ----
    {
        int waves  = (((Nn + 15) / 16)) * (256 / 64);  // 16x64 strip per wave
        int blocks = (waves + 7) / 8;                  // 8 waves per 256-thread block
        gemm_wmma_f32<<<blocks, 256, 0, stream>>>(x, Wl1, XL, Nn, 256, 8);
        gemm_wmma_f32<<<blocks, 256, 0, stream>>>(x, Wr1, XR, Nn, 256, 8);
    }

    // ---- layer 1 edge passes (H=4, C=64) ----
    {
        int blocks = (E2 + 7) / 8;   // 8 waves (edges) per block
        edge_alpha_kernel<4, 64><<<blocks, 256, 0, stream>>>(ei, ea, LA, XL, XR, We1, att1,
                                                             AL, AM, E, Nn);
        edge_agg_kernel<4, 64><<<blocks, 256, 0, stream>>>(ei, AL, AM, XL, AG, DN, E, Nn);
    }

    // ---- layer 1 finalize: /denom + b1, LN(g1,beta1), ELU -> h1 (in AG) ----
    finalize_kernel<4, 256><<<Nn, 256, 0, stream>>>(AG, DN, b1, g1, be1, Nn,
                                                    nullptr, nullptr, nullptr, 0);

    // ---- layer 2 node transforms via WMMA f32 (M=Nn, N=128, K=256) ----
    {
        int waves  = (((Nn + 15) / 16)) * (128 / 64);
        int blocks = (waves + 7) / 8;
        gemm_wmma_f32<<<blocks, 256, 0, stream>>>(AG, Wl2, XL, Nn, 128, 256);
        gemm_wmma_f32<<<blocks, 256, 0, stream>>>(AG, Wr2, XR, Nn, 128, 256);
    }

    // ---- reinit accumulators for layer 2 ----
    FILLF(AG, 0.0f, (long)Nn * 128);
    FILLF(AM, NINF, Nn);
    FILLF(DN, 0.0f, Nn);

    // ---- layer 2 edge passes (H=1, C=128) ----
    {
        int blocks = (E2 + 7) / 8;
        edge_alpha_kernel<1, 128><<<blocks, 256, 0, stream>>>(ei, ea, LA, XL, XR, We2, att2,
                                                              AL, AM, E, Nn);
        edge_agg_kernel<1, 128><<<blocks, 256, 0, stream>>>(ei, AL, AM, XL, AG, DN, E, Nn);
    }

    // ---- layer 2 finalize + fused graph mean-pool accumulation ----
    finalize_kernel<1, 128><<<Nn, 128, 0, stream>>>(AG, DN, b2, g2, be2, Nn,
                                                    batch, PO, CT, 1);

    // ---- final divide into d_out ----
    pool_div_kernel<<<(NG * 128 + 255) / 256, 256, 0, stream>>>(PO, CT, (float*)d_out, NG * 128);

#undef FILLF
}